// SO3Linear_42305427866200
// MI455X (gfx1250) — compile-verified
//
#include <hip/hip_runtime.h>

typedef float v2f __attribute__((ext_vector_type(2)));
typedef float v4f __attribute__((ext_vector_type(4)));
typedef float v8f __attribute__((ext_vector_type(8)));

#define B_TOT 16384
#define M_TOT 49
#define IN_F 128
#define OUT_F 128
#define ROWS_PER_BLOCK 128
#define LDS_STRIDE 132   // 128 + 4 floats pad -> conflict-free strided b64 reads

__global__ __launch_bounds__(256)
void so3_linear_wmma_f32(const float* __restrict__ x,
                         const float* __restrict__ w,
                         const float* __restrict__ bias,
                         float* __restrict__ out)
{
    extern __shared__ float lds[];
    float* Xs = lds;                                   // [128][132]
    float* Ws = lds + ROWS_PER_BLOCK * LDS_STRIDE;     // [128][132]

    const int bblk = blockIdx.x;   // 0..127 (batch tile)
    const int m    = blockIdx.y;   // 0..48
    // l(m): largest l with l*l <= m  (expand_index equivalent, computed scalar)
    int l = 0;
    while ((l + 1) * (l + 1) <= m) ++l;

    const int tid = threadIdx.x;
    const int b0  = bblk * ROWS_PER_BLOCK;

    // ---- stage X tile and W_l into padded LDS (coalesced float4) ----
    const float* wl = w + (size_t)l * OUT_F * IN_F;
    #pragma unroll 4
    for (int q = tid; q < ROWS_PER_BLOCK * (IN_F / 4); q += 256) {
        const int r  = q >> 5;     // row 0..127
        const int c4 = q & 31;     // float4 column 0..31
        v4f xv = *(const v4f*)(x + ((size_t)(b0 + r) * M_TOT + m) * IN_F + c4 * 4);
        *(v4f*)(Xs + r * LDS_STRIDE + c4 * 4) = xv;
        v4f wv = *(const v4f*)(wl + (size_t)r * IN_F + c4 * 4);
        *(v4f*)(Ws + r * LDS_STRIDE + c4 * 4) = wv;
    }
    __syncthreads();

    // ---- WMMA compute: each wave does 16 rows x 128 cols ----
    const int wave = tid >> 5;        // 0..7
    const int lane = tid & 31;
    const int half = lane >> 4;       // K-half selector per ISA fragment layout
    const int lc   = lane & 15;       // A: row-in-tile, B/C/D: column

    const float* Arow  = Xs + (wave * 16 + lc) * LDS_STRIDE + 2 * half;
    const float* Bbase = Ws + lc * LDS_STRIDE + 2 * half;

    v8f acc[8];
    const v8f zero = {0.f, 0.f, 0.f, 0.f, 0.f, 0.f, 0.f, 0.f};
    #pragma unroll
    for (int j = 0; j < 8; ++j) acc[j] = zero;

    #pragma unroll 4
    for (int t = 0; t < 32; ++t) {          // K steps of 4
        v2f a = *(const v2f*)(Arow + t * 4);
        #pragma unroll
        for (int j = 0; j < 8; ++j) {       // 8 N-tiles of 16
            v2f bf = *(const v2f*)(Bbase + j * 16 * LDS_STRIDE + t * 4);
            acc[j] = __builtin_amdgcn_wmma_f32_16x16x4_f32(
                false, a, false, bf, (short)0, acc[j], false, false);
        }
    }

    // ---- epilogue: bias on m==0 only, coalesced 64B half-wave stores ----
    #pragma unroll
    for (int j = 0; j < 8; ++j) {
        const int n  = j * 16 + lc;
        const float bv = (m == 0) ? bias[n] : 0.0f;
        const size_t base =
            ((size_t)(b0 + wave * 16 + 8 * half) * M_TOT + m) * OUT_F + n;
        #pragma unroll
        for (int r = 0; r < 8; ++r) {
            out[base + (size_t)r * (M_TOT * OUT_F)] = acc[j][r] + bv;
        }
    }
}

extern "C" void kernel_launch(void* const* d_in, const int* in_sizes, int n_in,
                              void* d_out, int out_size, void* d_ws, size_t ws_size,
                              hipStream_t stream) {
    const float* x    = (const float*)d_in[0];  // [16384,49,128] f32
    const float* w    = (const float*)d_in[1];  // [7,128,128]    f32
    const float* bias = (const float*)d_in[2];  // [1,1,128]      f32
    // d_in[3] = expand_index; l(m) is recomputed in-kernel.
    float* out = (float*)d_out;                 // [16384,49,128] f32

    const size_t shmem = 2u * ROWS_PER_BLOCK * LDS_STRIDE * sizeof(float); // 135168 B
    static_assert(2u * ROWS_PER_BLOCK * LDS_STRIDE * sizeof(float) <= 320u * 1024u,
                  "fits CDNA5 WGP LDS");

    // Allow >64KB dynamic LDS (CDNA5 WGP has 320KB).
    (void)hipFuncSetAttribute((const void*)so3_linear_wmma_f32,
                              hipFuncAttributeMaxDynamicSharedMemorySize,
                              (int)shmem);

    dim3 grid(B_TOT / ROWS_PER_BLOCK, M_TOT);
    so3_linear_wmma_f32<<<grid, 256, shmem, stream>>>(x, w, bias, out);
}